// MaskRCNN_61117384622358
// MI455X (gfx1250) — compile-verified
//
#include <hip/hip_runtime.h>
#include <cstdint>

// Problem constants (from reference): N, M, R, C, H, W = 2, 512, 8, 81, 56, 56
#define K_N  2
#define K_M  512
#define K_R  8
#define K_C  81
#define K_H  56
#define K_W  56
#define K_HW (K_H * K_W)      // 3136
#define K_MR (K_M * K_R)      // 4096

// Output sections, concatenated flat in reference return order (float elements)
#define ROIS_OFF 0
#define ROIS_SZ  (K_N * K_MR * 2 * 5)      // 81,920
#define CLS_OFF  (ROIS_OFF + ROIS_SZ)
#define CLS_SZ   (K_N * K_MR * 2 * K_C)    // 1,327,104
#define BBOX_OFF (CLS_OFF + CLS_SZ)
#define BBOX_SZ  (K_N * K_MR * 2 * 4)      // 65,536
#define MASK_OFF (BBOX_OFF + BBOX_SZ)      // 1,474,560

typedef float f4 __attribute__((ext_vector_type(4)));

// pos(n,m,r) = (IoU(proposal box, gt box) >= 0.5) ? 1 : 0, matching reference math.
__device__ __forceinline__ float iou_pos(const float* __restrict__ prop,
                                         const float* __restrict__ gtb,
                                         int n, int m, int r) {
    const float* p = prop + (size_t)(n * K_M + m) * 5;
    const float* g = gtb  + (size_t)(n * K_R + r) * 4;
    float a0 = p[1], a1 = p[2], a2 = p[3], a3 = p[4];
    float b0 = g[0], b1 = g[1], b2 = g[2], b3 = g[3];
    float ix1 = fmaxf(a0, b0), iy1 = fmaxf(a1, b1);
    float ix2 = fminf(a2, b2), iy2 = fminf(a3, b3);
    float iw = fmaxf(ix2 - ix1, 0.0f), ih = fmaxf(iy2 - iy1, 0.0f);
    float inter = iw * ih;
    float aa = (a2 - a0) * (a3 - a1);
    float ab = (b2 - b0) * (b3 - b1);
    float iou = inter / (aa + ab - inter + 1e-8f);
    return (iou >= 0.5f) ? 1.0f : 0.0f;
}

// ---------------------------------------------------------------------------
// Kernel 1: rois / cls_t / bbox_t (≈5.9 MB). One thread per output element,
// pos recomputed from cached inputs (all of proposals/gt_* live in L1/L2).
// ---------------------------------------------------------------------------
__global__ void small_targets_kernel(const float* __restrict__ prop,
                                     const float* __restrict__ gtc,
                                     const float* __restrict__ gtb,
                                     float* __restrict__ out) {
    int idx = blockIdx.x * blockDim.x + threadIdx.x;
    const int total = ROIS_SZ + CLS_SZ + BBOX_SZ;
    if (idx >= total) return;

    float val;
    int off;
    if (idx < ROIS_SZ) {
        int j  = idx % 5;
        int k  = (idx / 5) & 1;
        int mr = (idx / 10) % K_MR;
        int n  = idx / (10 * K_MR);
        int m = mr / K_R, r = mr % K_R;
        float pos = iou_pos(prop, gtb, n, m, r);
        float sel = k ? pos : (1.0f - pos);
        val = prop[(size_t)(n * K_M + m) * 5 + j] * sel;
        off = ROIS_OFF + idx;
    } else if (idx < ROIS_SZ + CLS_SZ) {
        int rel = idx - ROIS_SZ;
        int c  = rel % K_C;
        int k  = (rel / K_C) & 1;
        int mr = (rel / (2 * K_C)) % K_MR;
        int n  = rel / (2 * K_C * K_MR);
        int m = mr / K_R, r = mr % K_R;
        float pos = iou_pos(prop, gtb, n, m, r);
        float sel = k ? pos : (1.0f - pos);
        val = gtc[(size_t)(n * K_R + r) * K_C + c] * sel;
        off = CLS_OFF + rel;
    } else {
        int rel = idx - ROIS_SZ - CLS_SZ;
        int j  = rel % 4;
        int k  = (rel / 4) & 1;
        int mr = (rel / 8) % K_MR;
        int n  = rel / (8 * K_MR);
        int m = mr / K_R, r = mr % K_R;
        float pos = iou_pos(prop, gtb, n, m, r);
        float sel = k ? pos : (1.0f - pos);
        const float* g = gtb + (size_t)(n * K_R + r) * 4;
        float c;
        switch (j) {
            case 0:  c = (g[0] + g[2]) * 0.5f; break;   // cx
            case 1:  c = (g[1] + g[3]) * 0.5f; break;   // cy
            case 2:  c = g[2] - g[0];          break;   // w
            default: c = g[3] - g[1];          break;   // h
        }
        val = c * sel;
        off = BBOX_OFF + rel;
    }
    __builtin_nontemporal_store(val, out + off);
}

// ---------------------------------------------------------------------------
// Kernel 2: mask_t (205 MB of streaming stores — the bandwidth-bound part).
// Block = one (n, r) mask tile × 4 proposal rows m.
//   - Stage gt_masks[n,r] (12,544 B) into LDS with gfx1250 async global->LDS
//     b128 copies (ASYNCcnt path), overlap with IoU computation.
//   - Cache the tile in registers, then stream 4 m × {pos slot = data,
//     neg slot = zeros} with non-temporal b128 stores.
// ---------------------------------------------------------------------------
#define TPB        256
#define M_PER_BLK  4
#define NF4        (K_HW / 4)    // 784 float4 per tile; 784 = 3*256 + 16

__global__ void __launch_bounds__(TPB)
mask_targets_kernel(const float* __restrict__ prop,
                    const float* __restrict__ gtb,
                    const float* __restrict__ gtm,
                    float* __restrict__ out) {
    __shared__ __align__(16) float lds_mask[K_HW];
    __shared__ float lds_pos[M_PER_BLK];

    const int nr  = blockIdx.x;            // 0 .. N*R-1
    const int n   = nr / K_R;
    const int r   = nr % K_R;
    const int m0  = blockIdx.y * M_PER_BLK;
    const int tid = threadIdx.x;

    const float* src = gtm + (size_t)(n * K_R + r) * K_HW;

    // Async copy global -> LDS, 16 bytes per active lane per issue.
    // Low 32 bits of a generic pointer to __shared__ data are the LDS offset.
    const uint32_t lds_base = (uint32_t)(uintptr_t)(&lds_mask[0]);
    for (int i = tid; i < NF4; i += TPB) {
        uint32_t loff = lds_base + (uint32_t)(i * 16);
        uint64_t gaddr = (uint64_t)(uintptr_t)(src + i * 4);
        asm volatile("global_load_async_to_lds_b128 %0, %1, off"
                     :: "v"(loff), "v"(gaddr) : "memory");
    }

    // Overlap: compute the 4 pos flags while the async engine streams the tile.
    if (tid < M_PER_BLK) {
        lds_pos[tid] = iou_pos(prop, gtb, n, m0 + tid, r);
    }

    asm volatile("s_wait_asynccnt 0" ::: "memory");
    __syncthreads();

    // Cache this thread's slice of the tile in registers (3-4 float4).
    const f4* lm = (const f4*)lds_mask;
    f4 v0 = lm[tid];
    f4 v1 = lm[tid + 256];
    f4 v2 = lm[tid + 512];
    const bool has4 = (tid < (NF4 - 3 * TPB));   // tid < 16
    f4 v3 = has4 ? lm[tid + 768] : (f4){0.0f, 0.0f, 0.0f, 0.0f};
    const f4 zero = {0.0f, 0.0f, 0.0f, 0.0f};

    // Stream out 4 proposal rows × 2 slots with NT stores.
    for (int mi = 0; mi < M_PER_BLK; ++mi) {
        const int m = m0 + mi;
        const int kpos = (lds_pos[mi] >= 0.5f) ? 1 : 0;
        const size_t base =
            (size_t)MASK_OFF + (size_t)((n * K_MR + m * K_R + r) * 2) * K_HW;
        f4* dp = (f4*)(out + base + (size_t)kpos * K_HW);
        f4* dn = (f4*)(out + base + (size_t)(1 - kpos) * K_HW);

        __builtin_nontemporal_store(v0, dp + tid);
        __builtin_nontemporal_store(v1, dp + tid + 256);
        __builtin_nontemporal_store(v2, dp + tid + 512);
        if (has4) __builtin_nontemporal_store(v3, dp + tid + 768);

        __builtin_nontemporal_store(zero, dn + tid);
        __builtin_nontemporal_store(zero, dn + tid + 256);
        __builtin_nontemporal_store(zero, dn + tid + 512);
        if (has4) __builtin_nontemporal_store(zero, dn + tid + 768);
    }
}

extern "C" void kernel_launch(void* const* d_in, const int* in_sizes, int n_in,
                              void* d_out, int out_size, void* d_ws, size_t ws_size,
                              hipStream_t stream) {
    (void)in_sizes; (void)n_in; (void)out_size; (void)d_ws; (void)ws_size;
    const float* prop = (const float*)d_in[0];   // proposals  (N, M, 5)
    const float* gtc  = (const float*)d_in[1];   // gt_classes (N, R, C)
    const float* gtb  = (const float*)d_in[2];   // gt_bboxes  (N, R, 4)
    const float* gtm  = (const float*)d_in[3];   // gt_masks   (N, R, H, W)
    float* out = (float*)d_out;

    const int total_small = ROIS_SZ + CLS_SZ + BBOX_SZ;   // 1,474,560
    small_targets_kernel<<<(total_small + 255) / 256, 256, 0, stream>>>(
        prop, gtc, gtb, out);

    dim3 grid(K_N * K_R, K_M / M_PER_BLK);                // (16, 128) = 2048 blocks
    mask_targets_kernel<<<grid, TPB, 0, stream>>>(prop, gtb, gtm, out);
}